// NodeTypePredictor_33749853012398
// MI455X (gfx1250) — compile-verified
//
#include <hip/hip_runtime.h>
#include <math.h>

// ---------------------------------------------------------------------------
// 2-layer edge-featured GAT + MLP for MI455X (gfx1250, wave32).
// Dense GEMMs + classifier MLP run on V_WMMA_F32_16X16X4_F32 fed from LDS
// tiles staged with GLOBAL_LOAD_ASYNC_TO_LDS_B128 (ASYNCcnt) when the
// toolchain exposes the builtin (guarded; synchronous LDS fallback else).
// Edge phase is atomic segment ops (memory/atomic bound - roofline dominant).
// ---------------------------------------------------------------------------

typedef __attribute__((ext_vector_type(2))) float v2f;
typedef __attribute__((ext_vector_type(8))) float v8f;
typedef __attribute__((ext_vector_type(4))) int i32x4;

#define NEG_SLOPE 0.2f
#define F_IN 128
#define HDIM 64
#define FE 16
#define TOUT 32

// ---- async global->LDS staging (CDNA5 GLOBAL_LOAD_ASYNC_TO_LDS_B128) ------
#if defined(__has_builtin)
#if __has_builtin(__builtin_amdgcn_global_load_async_to_lds_b128) && \
    __has_builtin(__builtin_amdgcn_s_wait_asynccnt)
#define HAVE_ASYNC_LDS 1
#endif
#endif

__device__ __forceinline__ void g2l_b128(float* dst_lds, const float* src_g) {
#if defined(HAVE_ASYNC_LDS)
  // builtin signature (from toolchain): (V4i addrspace(1)*, V4i addrspace(3)*,
  //                                      imm int offset, imm int cpol)
  __builtin_amdgcn_global_load_async_to_lds_b128(
      (__attribute__((address_space(1))) i32x4*)src_g,
      (__attribute__((address_space(3))) i32x4*)dst_lds, 0, 0);
#else
  *(float4*)dst_lds = *(const float4*)src_g;
#endif
}

__device__ __forceinline__ void g2l_fence() {
#if defined(HAVE_ASYNC_LDS)
  __builtin_amdgcn_s_wait_asynccnt(0);
#endif
  __syncthreads();
}

__device__ __forceinline__ float lrelu(float x) { return x > 0.f ? x : NEG_SLOPE * x; }

__device__ __forceinline__ void atomicMaxF(float* addr, float val) {
  unsigned int* ua = reinterpret_cast<unsigned int*>(addr);
  unsigned int cur = __float_as_uint(*addr);
  while (__uint_as_float(cur) < val) {
    unsigned int prev = atomicCAS(ua, cur, __float_as_uint(val));
    if (prev == cur) break;
    cur = prev;
  }
}

// -------------------------- WMMA GEMM: C[M,64] = A[M,K] @ B[K,64] ----------
// 128 threads (4 waves); 64-row strip per block. A tile (64xK) and full B
// (Kx64) staged once into padded LDS (conflict-free A-frag b64 reads), then
// the whole K loop runs from LDS into f32 WMMA accumulators.
#define SA_LD (F_IN + 4)  // 132: bank shift of 4/row -> conflict-free a-frags
#define SB_LD 68          // 16B-aligned padded stride for B rows

__global__ void gat_wmma_gemm(const float* __restrict__ A, const float* __restrict__ B,
                              float* __restrict__ C, int M, int K) {
  __shared__ __align__(16) float sA[64 * SA_LD];
  __shared__ __align__(16) float sB[F_IN * SB_LD];

  const int tid = threadIdx.x;
  const int lane = tid & 31;
  const int wave = tid >> 5;
  const int mblk = blockIdx.x * 64;

  // stage A (64 x K) -- OOB rows clamp to row 0, zeroed at frag build
  const int kv = K >> 2;  // float4 per A row
  for (int idx = tid; idx < 64 * kv; idx += 128) {
    int r = idx / kv, c4 = idx % kv;
    int gr = mblk + r;
    g2l_b128(&sA[r * SA_LD + (c4 << 2)],
             A + (size_t)(gr < M ? gr : 0) * K + (c4 << 2));
  }
  // stage B (K x 64)
  for (int idx = tid; idx < K * 16; idx += 128) {
    int r = idx >> 4, c4 = idx & 15;
    g2l_b128(&sB[r * SB_LD + (c4 << 2)], B + (size_t)r * HDIM + (c4 << 2));
  }
  g2l_fence();

  // A frag (ISA 7.12.2): lane L holds M=L%16, K-pair {0,1} (L<16) or {2,3}.
  const int lm = lane & 15;
  const int lk = (lane < 16) ? 0 : 2;
  const int arow = mblk + wave * 16 + lm;
  const float rsc = (arow < M) ? 1.f : 0.f;  // zero OOB rows, no divergence
  const float* __restrict__ aL = &sA[(wave * 16 + lm) * SA_LD];

  v8f acc0 = {}, acc1 = {}, acc2 = {}, acc3 = {};
  for (int k0 = 0; k0 < K; k0 += 4) {
    v2f a;
    a.x = aL[k0 + lk] * rsc;
    a.y = aL[k0 + lk + 1] * rsc;
    const float* __restrict__ b0 = &sB[(k0 + lk) * SB_LD + lm];
    const float* __restrict__ b1 = b0 + SB_LD;
    v2f b;
    b.x = b0[0];  b.y = b1[0];
    acc0 = __builtin_amdgcn_wmma_f32_16x16x4_f32(false, a, false, b, (short)0, acc0, false, false);
    b.x = b0[16]; b.y = b1[16];
    acc1 = __builtin_amdgcn_wmma_f32_16x16x4_f32(false, a, false, b, (short)0, acc1, false, false);
    b.x = b0[32]; b.y = b1[32];
    acc2 = __builtin_amdgcn_wmma_f32_16x16x4_f32(false, a, false, b, (short)0, acc2, false, false);
    b.x = b0[48]; b.y = b1[48];
    acc3 = __builtin_amdgcn_wmma_f32_16x16x4_f32(false, a, false, b, (short)0, acc3, false, false);
  }

  // C/D layout: VGPR r -> M = r + 8*(lane>=16), N = lane%16 (+16 per tile)
  const int rbase = mblk + wave * 16 + ((lane >> 4) << 3);
#pragma unroll
  for (int r = 0; r < 8; ++r) {
    const int row = rbase + r;
    if (row < M) {
      float* __restrict__ crow = C + (size_t)row * HDIM + lm;
      crow[0]  = acc0[r];
      crow[16] = acc1[r];
      crow[32] = acc2[r];
      crow[48] = acc3[r];
    }
  }
}

// ------------------ fused WMMA classifier: relu(h@W1+b1)@W2+b2 -------------
#define SH_LD 68
#define SW_LD 36

__global__ void mlp_wmma_kernel(const float* __restrict__ h, const float* __restrict__ Wc1,
                                const float* __restrict__ bc1, const float* __restrict__ Wc2,
                                const float* __restrict__ bc2, float* __restrict__ out, int n) {
  __shared__ __align__(16) float sH[64 * SH_LD];
  __shared__ __align__(16) float sW1[HDIM * SW_LD];   // 64 x 32 padded
  __shared__ __align__(16) float sW2[TOUT * SW_LD];   // 32 x 32 padded
  __shared__ __align__(16) float sT[4 * 16 * SW_LD];  // per-wave 16 x 32 intermediate
  __shared__ float sb1[TOUT];
  __shared__ float sb2[TOUT];

  const int tid = threadIdx.x;
  const int lane = tid & 31;
  const int wave = tid >> 5;
  const int mblk = blockIdx.x * 64;

  for (int idx = tid; idx < 64 * 16; idx += 128) {      // h tile: 64 rows
    int r = idx >> 4, c4 = idx & 15;
    int gr = mblk + r;
    g2l_b128(&sH[r * SH_LD + (c4 << 2)],
             h + (size_t)(gr < n ? gr : 0) * HDIM + (c4 << 2));
  }
  for (int idx = tid; idx < HDIM * 8; idx += 128) {     // Wc1: 64 x 32
    int r = idx >> 3, c4 = idx & 7;
    g2l_b128(&sW1[r * SW_LD + (c4 << 2)], Wc1 + (size_t)r * TOUT + (c4 << 2));
  }
  for (int idx = tid; idx < TOUT * 8; idx += 128) {     // Wc2: 32 x 32
    int r = idx >> 3, c4 = idx & 7;
    g2l_b128(&sW2[r * SW_LD + (c4 << 2)], Wc2 + (size_t)r * TOUT + (c4 << 2));
  }
  if (tid < TOUT) sb1[tid] = bc1[tid];
  else if (tid < 2 * TOUT) sb2[tid - TOUT] = bc2[tid - TOUT];
  g2l_fence();

  const int lm = lane & 15;
  const int lk = (lane < 16) ? 0 : 2;
  const int hi = lane >> 4;
  const float* __restrict__ aL = &sH[(wave * 16 + lm) * SH_LD];

  // GEMM1: 16x64 @ 64x32 (2 tiles, 16 k-steps)
  v8f c0 = {}, c1 = {};
  for (int k0 = 0; k0 < HDIM; k0 += 4) {
    v2f a;
    a.x = aL[k0 + lk];
    a.y = aL[k0 + lk + 1];
    const float* __restrict__ b0 = &sW1[(k0 + lk) * SW_LD + lm];
    const float* __restrict__ b1 = b0 + SW_LD;
    v2f b;
    b.x = b0[0];  b.y = b1[0];
    c0 = __builtin_amdgcn_wmma_f32_16x16x4_f32(false, a, false, b, (short)0, c0, false, false);
    b.x = b0[16]; b.y = b1[16];
    c1 = __builtin_amdgcn_wmma_f32_16x16x4_f32(false, a, false, b, (short)0, c1, false, false);
  }

  // bias + relu in the accumulator layout; park into per-wave LDS tile so the
  // C-layout (row striped across lanes) becomes an A-layout (row per lane).
  float* __restrict__ tw = &sT[wave * 16 * SW_LD];
  const float bb0 = sb1[lm];
  const float bb1 = sb1[16 + lm];
#pragma unroll
  for (int r = 0; r < 8; ++r) {
    int tr = r + (hi << 3);
    tw[tr * SW_LD + lm]      = fmaxf(c0[r] + bb0, 0.f);
    tw[tr * SW_LD + 16 + lm] = fmaxf(c1[r] + bb1, 0.f);
  }
  __syncthreads();

  // GEMM2: 16x32 @ 32x32 (2 tiles, 8 k-steps)
  const float* __restrict__ a2 = &tw[lm * SW_LD];
  v8f d0 = {}, d1 = {};
  for (int k0 = 0; k0 < TOUT; k0 += 4) {
    v2f a;
    a.x = a2[k0 + lk];
    a.y = a2[k0 + lk + 1];
    const float* __restrict__ b0 = &sW2[(k0 + lk) * SW_LD + lm];
    const float* __restrict__ b1 = b0 + SW_LD;
    v2f b;
    b.x = b0[0];  b.y = b1[0];
    d0 = __builtin_amdgcn_wmma_f32_16x16x4_f32(false, a, false, b, (short)0, d0, false, false);
    b.x = b0[16]; b.y = b1[16];
    d1 = __builtin_amdgcn_wmma_f32_16x16x4_f32(false, a, false, b, (short)0, d1, false, false);
  }

  const float e0 = sb2[lm];
  const float e1 = sb2[16 + lm];
  const int rbase = mblk + wave * 16 + (hi << 3);
#pragma unroll
  for (int r = 0; r < 8; ++r) {
    const int row = rbase + r;
    if (row < n) {
      float* __restrict__ orow = out + (size_t)row * TOUT + lm;
      orow[0]  = d0[r] + e0;
      orow[16] = d1[r] + e1;
    }
  }
}

// -------------------------- edge/segment kernels ---------------------------
__global__ void zero_kernel(float* __restrict__ p, size_t n) {
  size_t i = (size_t)blockIdx.x * blockDim.x + threadIdx.x;
  if (i < n) p[i] = 0.f;
}

__global__ void edge_accum_kernel(const float* __restrict__ edge_attr, const int* __restrict__ dst,
                                  float* __restrict__ sumattr, float* __restrict__ deg, int E) {
  size_t gid = (size_t)blockIdx.x * blockDim.x + threadIdx.x;
  size_t e = gid >> 4;
  int f = (int)(gid & 15);
  if (e >= (size_t)E) return;
  int d = dst[e];
  atomicAdd(&sumattr[(size_t)d * FE + f], edge_attr[e * FE + f]);
  if (f == 0) atomicAdd(&deg[d], 1.0f);
}

__global__ void ve_kernel(const float* __restrict__ We, const float* __restrict__ ae,
                          float* __restrict__ ve) {
  int f = threadIdx.x;
  if (f >= FE) return;
  float s = 0.f;
  for (int j = 0; j < HDIM; ++j) s += We[f * HDIM + j] * ae[j];
  ve[f] = s;
}

__global__ void node_scal_kernel(const float* __restrict__ h, const float* __restrict__ a_s,
                                 const float* __restrict__ a_d, float* __restrict__ hs,
                                 float* __restrict__ hd, int n) {
  int i = blockIdx.x * blockDim.x + threadIdx.x;
  if (i >= n) return;
  const float* __restrict__ r = h + (size_t)i * HDIM;
  float s = 0.f, d = 0.f;
#pragma unroll 8
  for (int f = 0; f < HDIM; ++f) { float v = r[f]; s += v * a_s[f]; d += v * a_d[f]; }
  hs[i] = s; hd[i] = d;
}

__global__ void he_kernel(const float* __restrict__ edge_attr, const float* __restrict__ ve,
                          float* __restrict__ ebuf, int E) {
  int e = blockIdx.x * blockDim.x + threadIdx.x;
  if (e >= E) return;
  const float* __restrict__ ea = edge_attr + (size_t)e * FE;
  float s = 0.f;
#pragma unroll
  for (int f = 0; f < FE; ++f) s += ea[f] * ve[f];
  ebuf[e] = s;
}

__global__ void loop_init_kernel(const float* __restrict__ hs, const float* __restrict__ hd,
                                 const float* __restrict__ sumattr, const float* __restrict__ deg,
                                 const float* __restrict__ ve, float* __restrict__ aloop,
                                 float* __restrict__ m, int n) {
  int i = blockIdx.x * blockDim.x + threadIdx.x;
  if (i >= n) return;
  float dg = fmaxf(deg[i], 1.0f);
  float lh = 0.f;
#pragma unroll
  for (int f = 0; f < FE; ++f) lh += sumattr[(size_t)i * FE + f] * ve[f];
  lh /= dg;
  float a = lrelu(hs[i] + hd[i] + lh);
  aloop[i] = a;
  m[i] = a;
}

__global__ void edge_max_kernel(const int* __restrict__ src, const int* __restrict__ dst,
                                const float* __restrict__ hs, const float* __restrict__ hd,
                                float* __restrict__ ebuf, float* __restrict__ m, int E) {
  int e = blockIdx.x * blockDim.x + threadIdx.x;
  if (e >= E) return;
  int d = dst[e];
  float a = lrelu(hs[src[e]] + hd[d] + ebuf[e]);
  ebuf[e] = a;
  atomicMaxF(&m[d], a);
}

__global__ void z_init_kernel(const float* __restrict__ aloop, const float* __restrict__ m,
                              float* __restrict__ z, int n) {
  int i = blockIdx.x * blockDim.x + threadIdx.x;
  if (i >= n) return;
  z[i] = __expf(aloop[i] - m[i]);
}

__global__ void edge_exp_kernel(const int* __restrict__ dst, float* __restrict__ ebuf,
                                const float* __restrict__ m, float* __restrict__ z, int E) {
  int e = blockIdx.x * blockDim.x + threadIdx.x;
  if (e >= E) return;
  int d = dst[e];
  float p = __expf(ebuf[e] - m[d]);
  ebuf[e] = p;
  atomicAdd(&z[d], p);
}

__global__ void agg_init_kernel(const float* __restrict__ h, const float* __restrict__ aloop,
                                const float* __restrict__ m, const float* __restrict__ z,
                                float* __restrict__ out, int n) {
  size_t gid = (size_t)blockIdx.x * blockDim.x + threadIdx.x;
  size_t i = gid >> 6;
  int f = (int)(gid & 63);
  if (i >= (size_t)n) return;
  float w = __expf(aloop[i] - m[i]) / z[i];
  out[i * HDIM + f] = w * h[i * HDIM + f];
}

__global__ void edge_scatter_kernel(const int* __restrict__ src, const int* __restrict__ dst,
                                    const float* __restrict__ ebuf, const float* __restrict__ z,
                                    const float* __restrict__ h, float* __restrict__ out, int E) {
  size_t gid = (size_t)blockIdx.x * blockDim.x + threadIdx.x;
  size_t e = gid >> 6;
  int f = (int)(gid & 63);
  if (e >= (size_t)E) return;
  int d = dst[e];
  float w = ebuf[e] / z[d];
  atomicAdd(&out[(size_t)d * HDIM + f], w * h[(size_t)src[e] * HDIM + f]);
}

__global__ void bias_relu_kernel(float* __restrict__ x, const float* __restrict__ b, int n) {
  size_t gid = (size_t)blockIdx.x * blockDim.x + threadIdx.x;
  size_t i = gid >> 6;
  int f = (int)(gid & 63);
  if (i >= (size_t)n) return;
  float v = x[i * HDIM + f] + b[f];
  x[i * HDIM + f] = fmaxf(v, 0.f);
}

// ---------------------------------------------------------------------------
static inline unsigned nblk(size_t n, int b) { return (unsigned)((n + (size_t)b - 1) / (size_t)b); }

static void run_gat_layer(const float* X, int K, const float* W, const float* a_s,
                          const float* a_d, const float* We, const float* ae, const float* b,
                          const float* edge_attr, const int* src, const int* dst,
                          float* F0, float* F1, float* hs, float* hd, float* aloop,
                          float* m, float* z, float* deg, float* sumattr, float* ebuf,
                          float* ve, int N, int E, hipStream_t stream) {
  ve_kernel<<<1, FE, 0, stream>>>(We, ae, ve);
  gat_wmma_gemm<<<nblk((size_t)N, 64), 128, 0, stream>>>(X, W, F0, N, K);
  node_scal_kernel<<<nblk((size_t)N, 256), 256, 0, stream>>>(F0, a_s, a_d, hs, hd, N);
  he_kernel<<<nblk((size_t)E, 256), 256, 0, stream>>>(edge_attr, ve, ebuf, E);
  loop_init_kernel<<<nblk((size_t)N, 256), 256, 0, stream>>>(hs, hd, sumattr, deg, ve, aloop, m, N);
  edge_max_kernel<<<nblk((size_t)E, 256), 256, 0, stream>>>(src, dst, hs, hd, ebuf, m, E);
  z_init_kernel<<<nblk((size_t)N, 256), 256, 0, stream>>>(aloop, m, z, N);
  edge_exp_kernel<<<nblk((size_t)E, 256), 256, 0, stream>>>(dst, ebuf, m, z, E);
  agg_init_kernel<<<nblk((size_t)N * HDIM, 256), 256, 0, stream>>>(F0, aloop, m, z, F1, N);
  edge_scatter_kernel<<<nblk((size_t)E * HDIM, 256), 256, 0, stream>>>(src, dst, ebuf, z, F0, F1, E);
  bias_relu_kernel<<<nblk((size_t)N * HDIM, 256), 256, 0, stream>>>(F1, b, N);
}

extern "C" void kernel_launch(void* const* d_in, const int* in_sizes, int n_in,
                              void* d_out, int out_size, void* d_ws, size_t ws_size,
                              hipStream_t stream) {
  const float* x         = (const float*)d_in[0];
  const int*   ei        = (const int*)d_in[1];
  const float* edge_attr = (const float*)d_in[2];
  const float* W1  = (const float*)d_in[3];
  const float* as1 = (const float*)d_in[4];
  const float* ad1 = (const float*)d_in[5];
  const float* We1 = (const float*)d_in[6];
  const float* ae1 = (const float*)d_in[7];
  const float* b1  = (const float*)d_in[8];
  const float* W2  = (const float*)d_in[9];
  const float* as2 = (const float*)d_in[10];
  const float* ad2 = (const float*)d_in[11];
  const float* We2 = (const float*)d_in[12];
  const float* ae2 = (const float*)d_in[13];
  const float* b2  = (const float*)d_in[14];
  const float* Wc1 = (const float*)d_in[15];
  const float* bc1 = (const float*)d_in[16];
  const float* Wc2 = (const float*)d_in[17];
  const float* bc2 = (const float*)d_in[18];

  const int N = in_sizes[0] / F_IN;
  const int E = in_sizes[1] / 2;
  const int* src = ei;
  const int* dst = ei + E;

  // workspace carve-up (floats)
  float* ws = (float*)d_ws;
  size_t N64 = (size_t)N * HDIM;
  float* F0      = ws;                         // pre-aggregation features h
  float* F1      = F0 + N64;                   // aggregation output / next input
  float* hs      = F1 + N64;
  float* hd      = hs + N;
  float* aloop   = hd + N;
  float* m       = aloop + N;
  float* z       = m + N;
  float* deg     = z + N;
  float* sumattr = deg + N;                    // N * FE
  float* ebuf    = sumattr + (size_t)N * FE;   // E (he -> alpha -> p, in place)
  float* ve      = ebuf + E;                   // 16

  // degree + incoming edge-attr sums (used by both layers' self-loops)
  zero_kernel<<<nblk((size_t)N * (FE + 1), 256), 256, 0, stream>>>(deg, (size_t)N * (FE + 1));
  edge_accum_kernel<<<nblk((size_t)E * FE, 256), 256, 0, stream>>>(edge_attr, dst, sumattr, deg, E);

  // layer 1: x[N,128] -> F1 = relu(agg + b1)
  run_gat_layer(x, F_IN, W1, as1, ad1, We1, ae1, b1, edge_attr, src, dst,
                F0, F1, hs, hd, aloop, m, z, deg, sumattr, ebuf, ve, N, E, stream);
  // layer 2: F1[N,64] -> F1 = relu(agg + b2)  (F0 reused for h2)
  run_gat_layer(F1, HDIM, W2, as2, ad2, We2, ae2, b2, edge_attr, src, dst,
                F0, F1, hs, hd, aloop, m, z, deg, sumattr, ebuf, ve, N, E, stream);

  // classifier (fully WMMA, fused two-GEMM kernel)
  mlp_wmma_kernel<<<nblk((size_t)N, 64), 128, 0, stream>>>(F1, Wc1, bc1, Wc2, bc2,
                                                           (float*)d_out, N);
}